// HyperbolicMultiHeadAttention_75660143886519
// MI455X (gfx1250) — compile-verified
//
#include <hip/hip_runtime.h>
#include <hip/hip_bf16.h>

typedef float v2f __attribute__((ext_vector_type(2)));
typedef float v8f __attribute__((ext_vector_type(8)));
typedef int   vi4 __attribute__((vector_size(16)));

#define S_LEN   2048
#define D_DIM   512
#define NHEAD   8
#define DK      64
#define NROWS   4096                 // B * S
#define MAXN_C  0.99999f             // 1 - 1e-5
#define ACLIP_C (1.0f - 1e-7f)
#define EPS_C   1e-15f

__device__ __forceinline__ float artanh_c(float x) {
    x = fminf(fmaxf(x, -ACLIP_C), ACLIP_C);
    return 0.5f * __logf((1.0f + x) / (1.0f - x));
}
__device__ __forceinline__ float tanh_c(float x) {
    float e = __expf(fminf(2.0f * x, 30.0f));
    return (e - 1.0f) / (e + 1.0f);
}
// D = A(16x4) * B(4x16) + C, fp32 WMMA (wave32)
__device__ __forceinline__ v8f wmma4(v2f a, v2f b, v8f c) {
    return __builtin_amdgcn_wmma_f32_16x16x4_f32(false, a, false, b, (short)0, c,
                                                 false, false);
}
// sum across a 16-lane half-wave (masks < 16 stay inside the half)
__device__ __forceinline__ float hred16(float v) {
    v += __shfl_xor(v, 1); v += __shfl_xor(v, 2);
    v += __shfl_xor(v, 4); v += __shfl_xor(v, 8);
    return v;
}
// full wave32 all-reduce
__device__ __forceinline__ float red32(float v) {
    v = hred16(v); v += __shfl_xor(v, 16); return v;
}

// ---- CDNA5 async global->LDS (ASYNCcnt) with safe fallback -----------------
// Enable only if BOTH the DMA builtin and its wait builtin exist, so we never
// issue ASYNCcnt-tracked loads that we cannot wait on.
#if defined(__HIP_DEVICE_COMPILE__) && \
    __has_builtin(__builtin_amdgcn_global_load_async_to_lds_b128) && \
    __has_builtin(__builtin_amdgcn_s_wait_asynccnt)
#define HAS_ASYNC_LDS 1
#else
#define HAS_ASYNC_LDS 0
#endif

__device__ __forceinline__ void async_copy_b128(const float* g, float* l) {
#if HAS_ASYNC_LDS
    typedef __attribute__((address_space(1))) vi4* gp_t;   // global int4*
    typedef __attribute__((address_space(3))) vi4* lp_t;   // LDS int4*
    __builtin_amdgcn_global_load_async_to_lds_b128((gp_t)g, (lp_t)l, 0, 0);
#else
    *(float4*)l = *(const float4*)g;
#endif
}
__device__ __forceinline__ void async_wait0() {
#if HAS_ASYNC_LDS
    __builtin_amdgcn_s_wait_asynccnt(0);
#endif
}

// ---------------------------------------------------------------------------
// Y = project(mobius_add(mobius_matvec(W, X), b)) for a 16-row stripe.
// 512 threads = 16 waves; wave w owns columns [32w, 32w+32) as two 16x16 tiles.
// ---------------------------------------------------------------------------
__global__ __launch_bounds__(512)
void k_mobius_linear(const float* __restrict__ X, const float* __restrict__ W,
                     const float* __restrict__ bias, float* __restrict__ Y) {
    __shared__ float sX[16 * D_DIM];          // 32 KB stripe of X
    __shared__ float sxn2[16], smx2[16], sxy[16], su2[16];
    __shared__ float sby2;

    const int tid  = threadIdx.x;
    const int wave = tid >> 5;
    const int lane = tid & 31;
    const int row0 = blockIdx.x * 16;
    const int mrow  = lane & 15;
    const int khalf = (lane >> 4) * 2;        // 0 or 2
    const int n0    = wave * 32;

    if (tid < 16) { sxn2[tid] = 0.f; smx2[tid] = 0.f; sxy[tid] = 0.f; su2[tid] = 0.f; }
    if (tid == 16) sby2 = 0.f;

    // stage X stripe: 16 row-major-adjacent rows = 8192 contiguous floats.
    // CDNA5 async DMA path (global_load_async_to_lds_b128, ASYNCcnt).
    {
        const float* gsrc = X + (size_t)row0 * D_DIM;
        #pragma unroll
        for (int t = 0; t < 4; ++t) {
            int e4 = tid + t * 512;           // float4 index, 2048 total
            async_copy_b128(gsrc + e4 * 4, &sX[e4 * 4]);
        }
        async_wait0();
    }
    // ||b||^2 (wave-reduced, few LDS atomics)
    {
        float bv = bias[tid];
        float p  = red32(bv * bv);
        if (lane == 0) atomicAdd(&sby2, p);
    }
    __syncthreads();

    // ||x||^2 per row (wave w handles row w)
    {
        float s = 0.f;
        for (int c = lane; c < D_DIM; c += 32) {
            float v = sX[wave * D_DIM + c];
            s += v * v;
        }
        s = red32(s);
        if (lane == 0) sxn2[wave] = s;
    }

    // ---- GEMM: mx = x @ W^T via V_WMMA_F32_16X16X4_F32 ----
    v8f acc0 = {}; v8f acc1 = {};
    const float* w0p = &W[(size_t)(n0 + mrow) * D_DIM + khalf];
    const float* w1p = &W[(size_t)(n0 + 16 + mrow) * D_DIM + khalf];
    for (int k = 0; k < D_DIM; k += 4) {
        if ((k & 63) == 0 && k + 64 < D_DIM) {
            __builtin_prefetch(w0p + k + 64, 0, 3);
            __builtin_prefetch(w1p + k + 64, 0, 3);
        }
        v2f a  = *(const v2f*)&sX[mrow * D_DIM + k + khalf];
        v2f b0 = *(const v2f*)(w0p + k);
        v2f b1 = *(const v2f*)(w1p + k);
        acc0 = wmma4(a, b0, acc0);
        acc1 = wmma4(a, b1, acc1);
    }

    // ||mx||^2 per row: C/D layout -> vgpr r holds rows r (lanes 0-15) and r+8 (16-31)
    #pragma unroll
    for (int r = 0; r < 8; ++r) {
        int rr = r + ((lane >> 4) << 3);
        float p = hred16(acc0[r] * acc0[r] + acc1[r] * acc1[r]);
        if ((lane & 15) == 0) atomicAdd(&smx2[rr], p);
    }
    __syncthreads();

    const float bv0 = bias[n0 + (lane & 15)];
    const float bv1 = bias[n0 + 16 + (lane & 15)];

    // mobius_matvec epilogue + <res,b> reduction
    #pragma unroll
    for (int r = 0; r < 8; ++r) {
        int rr = r + ((lane >> 4) << 3);
        float xn  = sqrtf(fmaxf(sxn2[rr], EPS_C));
        float mxn = sqrtf(fmaxf(smx2[rr], EPS_C));
        float t   = tanh_c(mxn / xn * artanh_c(xn));
        float sc  = t / mxn;
        float r0 = sc * acc0[r], r1 = sc * acc1[r];
        acc0[r] = r0; acc1[r] = r1;
        float p = hred16(r0 * bv0 + r1 * bv1);
        if ((lane & 15) == 0) atomicAdd(&sxy[rr], p);
    }
    __syncthreads();

    // mobius_add + ||u||^2 reduction
    #pragma unroll
    for (int r = 0; r < 8; ++r) {
        int rr = r + ((lane >> 4) << 3);
        float xn  = sqrtf(fmaxf(sxn2[rr], EPS_C));
        float mxn = sqrtf(fmaxf(smx2[rr], EPS_C));
        float t   = tanh_c(mxn / xn * artanh_c(xn));
        float x2  = t * t;                       // ||res||^2 = tanh^2 exactly
        float y2  = sby2;
        float xy  = sxy[rr];
        float Ac  = 1.0f + 2.0f * xy + y2;
        float Bc  = 1.0f - x2;
        float den = fmaxf(1.0f + 2.0f * xy + x2 * y2, EPS_C);
        float u0 = (Ac * acc0[r] + Bc * bv0) / den;
        float u1 = (Ac * acc1[r] + Bc * bv1) / den;
        acc0[r] = u0; acc1[r] = u1;
        float p = hred16(u0 * u0 + u1 * u1);
        if ((lane & 15) == 0) atomicAdd(&su2[rr], p);
    }
    __syncthreads();

    // project + store
    #pragma unroll
    for (int r = 0; r < 8; ++r) {
        int rr = r + ((lane >> 4) << 3);
        float un = sqrtf(fmaxf(su2[rr], EPS_C));
        float f  = (un > MAXN_C) ? (MAXN_C / un) : 1.0f;
        size_t base = (size_t)(row0 + rr) * D_DIM;
        Y[base + n0 + (lane & 15)]      = f * acc0[r];
        Y[base + n0 + 16 + (lane & 15)] = f * acc1[r];
    }
}

// ---------------------------------------------------------------------------
// per-(row,head): q2 = ||Q_h||^2, k2 = ||K_h||^2, V_h <- logmap0(V_h) in place
// ---------------------------------------------------------------------------
__global__ __launch_bounds__(256)
void k_head_stats(const float* __restrict__ Qp, const float* __restrict__ Kp,
                  float* __restrict__ Vp, float* __restrict__ q2,
                  float* __restrict__ k2) {
    const int wave = threadIdx.x >> 5;
    const int lane = threadIdx.x & 31;
    const int task = blockIdx.x * 8 + wave;   // 0 .. 32767
    const int row  = task >> 3;               // global row (b*S + s)
    const int h    = task & 7;
    const int b    = row >> 11;
    const int srow = row & 2047;
    const size_t base = (size_t)row * D_DIM + h * DK;
    const int sidx = ((b << 3) + h) * S_LEN + srow;

    float qa = Qp[base + lane], qb = Qp[base + lane + 32];
    float sq = red32(qa * qa + qb * qb);
    if (lane == 0) q2[sidx] = sq;

    float ka = Kp[base + lane], kb = Kp[base + lane + 32];
    float sk = red32(ka * ka + kb * kb);
    if (lane == 0) k2[sidx] = sk;

    float va = Vp[base + lane], vb = Vp[base + lane + 32];
    float vn2 = red32(va * va + vb * vb);
    float vn  = sqrtf(fmaxf(vn2, EPS_C));
    float f   = artanh_c(vn) / vn;
    Vp[base + lane]      = f * va;
    Vp[base + lane + 32] = f * vb;
}

__global__ void k_zero(float* __restrict__ p, int n) {
    int i = blockIdx.x * 256 + threadIdx.x;
    if (i < n) p[i] = 0.f;
}

// ---------------------------------------------------------------------------
// scores: p = exp(-dist(q,k)/sqrt(dk)), unnormalized, + row sums.
// 8 waves, each wave = two 16x16 j-tiles; block covers 16 rows x 256 cols.
// ---------------------------------------------------------------------------
__global__ __launch_bounds__(256)
void k_scores(const float* __restrict__ Qp, const float* __restrict__ Kp,
              const float* __restrict__ q2g, const float* __restrict__ k2g,
              float* __restrict__ attn, float* __restrict__ rowsum) {
    __shared__ float sQ[16 * DK];            // 4 KB
    const int tid  = threadIdx.x;
    const int wave = tid >> 5;
    const int lane = tid & 31;
    const int bh = blockIdx.z;               // 0..15
    const int b  = bh >> 3, h = bh & 7;
    const int i0 = blockIdx.y * 16;
    const int j0 = blockIdx.x * 256 + wave * 32;
    const size_t bS = (size_t)b * S_LEN;
    const int mrow  = lane & 15;
    const int khalf = (lane >> 4) * 2;

    // stage Q tile (16 x 64) via async DMA: one b128 per thread
    {
        int r = tid >> 4, c = (tid & 15) * 4;    // 16 float4 per row
        async_copy_b128(&Qp[(bS + i0 + r) * D_DIM + h * DK + c], &sQ[r * DK + c]);
        async_wait0();
    }
    __syncthreads();

    v8f acc0 = {}; v8f acc1 = {};
    const float* k0p = &Kp[(bS + j0 + mrow) * D_DIM + h * DK + khalf];
    const float* k1p = &Kp[(bS + j0 + 16 + mrow) * D_DIM + h * DK + khalf];
    #pragma unroll
    for (int k = 0; k < DK; k += 4) {
        v2f a  = *(const v2f*)&sQ[mrow * DK + k + khalf];
        v2f b0 = *(const v2f*)(k0p + k);
        v2f b1 = *(const v2f*)(k1p + k);
        acc0 = wmma4(a, b0, acc0);
        acc1 = wmma4(a, b1, acc1);
    }

    const float y2a = k2g[bh * S_LEN + j0 + (lane & 15)];
    const float y2b = k2g[bh * S_LEN + j0 + 16 + (lane & 15)];

    #pragma unroll
    for (int r = 0; r < 8; ++r) {
        int rr = r + ((lane >> 4) << 3);
        float x2 = q2g[bh * S_LEN + i0 + rr];
        float xy0 = acc0[r], xy1 = acc1[r];

        float Ac  = 1.0f - 2.0f * xy0 + y2a;
        float Bc  = 1.0f - x2;
        float n2  = fmaxf(Ac * Ac * x2 + Bc * Bc * y2a - 2.0f * Ac * Bc * xy0, 0.0f);
        float den = fmaxf(1.0f - 2.0f * xy0 + x2 * y2a, EPS_C);
        float p0  = __expf(-0.25f * artanh_c(sqrtf(n2) / den)); // 2*artanh/8

        Ac  = 1.0f - 2.0f * xy1 + y2b;
        n2  = fmaxf(Ac * Ac * x2 + Bc * Bc * y2b - 2.0f * Ac * Bc * xy1, 0.0f);
        den = fmaxf(1.0f - 2.0f * xy1 + x2 * y2b, EPS_C);
        float p1 = __expf(-0.25f * artanh_c(sqrtf(n2) / den));

        size_t arow = ((size_t)bh * S_LEN + i0 + rr) * S_LEN;
        attn[arow + j0 + (lane & 15)]      = p0;
        attn[arow + j0 + 16 + (lane & 15)] = p1;

        float pr = hred16(p0 + p1);
        if ((lane & 15) == 0) atomicAdd(&rowsum[bh * S_LEN + i0 + rr], pr);
    }
}

__global__ void k_attn_norm(float4* __restrict__ attn4,
                            const float* __restrict__ rowsum) {
    size_t i = (size_t)blockIdx.x * 256 + threadIdx.x;   // < 16M
    size_t e = i * 4;
    float inv = 1.0f / rowsum[e >> 11];
    float4 v = attn4[i];
    v.x *= inv; v.y *= inv; v.z *= inv; v.w *= inv;
    attn4[i] = v;
}

// ---------------------------------------------------------------------------
// mid = attn @ logmap0(V); out = expmap0(mid), head-interleaved layout.
// 8 waves split K=2048 into 256-chunks; each wave does 4 d-tiles (dk=64).
// ---------------------------------------------------------------------------
__global__ __launch_bounds__(256)
void k_attnv(const float* __restrict__ attn, const float* __restrict__ LV,
             float* __restrict__ OutH) {
    __shared__ float sAcc[4 * 8 * 32];       // 4 KB cross-wave partials
    __shared__ float sun2[16];
    const int tid  = threadIdx.x;
    const int wave = tid >> 5;
    const int lane = tid & 31;
    const int bh = blockIdx.y;               // 0..15
    const int b  = bh >> 3, h = bh & 7;
    const int i0 = blockIdx.x * 16;
    const size_t bS = (size_t)b * S_LEN;
    const int khalf = (lane >> 4) * 2;

    #pragma unroll
    for (int t = 0; t < 4; ++t) sAcc[t * 256 + tid] = 0.f;
    if (tid < 16) sun2[tid] = 0.f;
    __syncthreads();

    v8f acc[4] = {v8f{}, v8f{}, v8f{}, v8f{}};
    const float* arow = attn + ((size_t)bh * S_LEN + i0 + (lane & 15)) * S_LEN;
    const float* lvb  = LV + bS * D_DIM + h * DK + (lane & 15);
    const int kbeg = wave * 256, kend = kbeg + 256;
    for (int kk = kbeg; kk < kend; kk += 4) {
        v2f a = *(const v2f*)(arow + kk + khalf);
        #pragma unroll
        for (int t = 0; t < 4; ++t) {
            v2f bb;
            bb.x = lvb[(size_t)(kk + khalf) * D_DIM + t * 16];
            bb.y = lvb[(size_t)(kk + khalf + 1) * D_DIM + t * 16];
            acc[t] = wmma4(a, bb, acc[t]);
        }
    }
    #pragma unroll
    for (int t = 0; t < 4; ++t)
        #pragma unroll
        for (int r = 0; r < 8; ++r)
            atomicAdd(&sAcc[t * 256 + r * 32 + lane], acc[t][r]);
    __syncthreads();

    if (wave < 4) {
        #pragma unroll
        for (int r = 0; r < 8; ++r) {
            int rr = r + ((lane >> 4) << 3);
            float v = sAcc[wave * 256 + r * 32 + lane];
            float p = hred16(v * v);
            if ((lane & 15) == 0) atomicAdd(&sun2[rr], p);
        }
    }
    __syncthreads();

    if (wave < 4) {
        #pragma unroll
        for (int r = 0; r < 8; ++r) {
            int rr = r + ((lane >> 4) << 3);
            float mid = sAcc[wave * 256 + r * 32 + lane];
            float un  = sqrtf(fmaxf(sun2[rr], EPS_C));
            float f   = tanh_c(un) / un;   // expmap0 scale
            OutH[(bS + i0 + rr) * D_DIM + h * DK + wave * 16 + (lane & 15)] = f * mid;
        }
    }
}

// ---------------------------------------------------------------------------
extern "C" void kernel_launch(void* const* d_in, const int* in_sizes, int n_in,
                              void* d_out, int out_size, void* d_ws, size_t ws_size,
                              hipStream_t stream) {
    (void)in_sizes; (void)n_in; (void)out_size; (void)ws_size;
    const float* query = (const float*)d_in[0];
    const float* key   = (const float*)d_in[1];
    const float* value = (const float*)d_in[2];
    const float* Wq = (const float*)d_in[3];  const float* bq = (const float*)d_in[4];
    const float* Wk = (const float*)d_in[5];  const float* bk = (const float*)d_in[6];
    const float* Wv = (const float*)d_in[7];  const float* bv = (const float*)d_in[8];
    const float* Wo = (const float*)d_in[9];  const float* bo = (const float*)d_in[10];

    float* out  = (float*)d_out;                      // (B,S,D) = 2,097,152
    float* attn = out + (size_t)NROWS * D_DIM;        // (B,H,S,S) = 67,108,864

    float* ws = (float*)d_ws;
    const size_t MAT = (size_t)NROWS * D_DIM;         // 2,097,152 floats
    float* Qp = ws;                                   // projected Q (head-interleaved)
    float* Kp = ws + MAT;
    float* Vp = ws + 2 * MAT;                         // becomes logmap0(V) in place
    float* q2 = ws + 3 * MAT;                         // 32768
    float* k2 = q2 + 32768;
    float* rowsum = k2 + 32768;                       // 32768
    float* OutH = Qp;                                 // reuse after scores pass

    // hyperbolic projections (WMMA GEMM + fused epilogue)
    k_mobius_linear<<<NROWS / 16, 512, 0, stream>>>(query, Wq, bq, Qp);
    k_mobius_linear<<<NROWS / 16, 512, 0, stream>>>(key,   Wk, bk, Kp);
    k_mobius_linear<<<NROWS / 16, 512, 0, stream>>>(value, Wv, bv, Vp);

    k_zero<<<32768 / 256, 256, 0, stream>>>(rowsum, 32768);
    k_head_stats<<<NROWS * NHEAD / 8, 256, 0, stream>>>(Qp, Kp, Vp, q2, k2);

    // scores -> unnormalized softmax + row sums
    k_scores<<<dim3(S_LEN / 256, S_LEN / 16, 16), 256, 0, stream>>>(
        Qp, Kp, q2, k2, attn, rowsum);

    // in-place normalize (final attn output)
    k_attn_norm<<<(unsigned)((size_t)16 * S_LEN * S_LEN / 4 / 256), 256, 0, stream>>>(
        (float4*)attn, rowsum);

    // gyromidpoint: attn @ logmap0(V), then expmap0
    k_attnv<<<dim3(S_LEN / 16, 16), 256, 0, stream>>>(attn, Vp, OutH);

    // output projection
    k_mobius_linear<<<NROWS / 16, 512, 0, stream>>>(OutH, Wo, bo, out);
}